// PillarFeatureNet_446676599108
// MI455X (gfx1250) — compile-verified
//
#include <hip/hip_runtime.h>

// PillarFeatureNet forward for MI455X (gfx1250), wave32 + V_WMMA_F32_16X16X4_F32.
//
// Pass A: feats build + WMMA GEMM + deterministic per-block partial sum/sumsq
// Pass B: finalize BN stats -> per-unit scale/bias
// Pass C: recompute feats + WMMA GEMM + affine + relu + max over points -> out
//
// Two-pass recompute avoids materializing x [P,N,64] (328 MB) -> ~51 MB total HBM traffic.

#define PILLARS 40000
#define NPTS    32
#define CIN     4
#define UNITS   64
#define KPAD    12           // K = 9 padded to 12 (3 wmma k-steps of 4)
#define WAVES_PER_BLOCK 8
#define BLOCK   (WAVES_PER_BLOCK * 32)
#define NBLK    ((PILLARS + WAVES_PER_BLOCK - 1) / WAVES_PER_BLOCK)   // 5000

typedef float v2f __attribute__((ext_vector_type(2)));
typedef float v8f __attribute__((ext_vector_type(8)));

// Build the 9-dim (padded to 12) masked feature row for this lane's point and
// store it into this wave's LDS staging area.
__device__ __forceinline__ void pillar_feats_to_lds(
    const float* __restrict__ features, const int* __restrict__ num_voxels,
    const int* __restrict__ coors, int p, int lane, float* __restrict__ frow)
{
    const float4 f = *reinterpret_cast<const float4*>(
        features + ((size_t)p * NPTS + lane) * CIN);

    // Unmasked sum over ALL 32 points (matches reference), divided by num_voxels.
    float sx = f.x, sy = f.y, sz = f.z;
#pragma unroll
    for (int m = 16; m >= 1; m >>= 1) {
        sx += __shfl_xor(sx, m, 32);
        sy += __shfl_xor(sy, m, 32);
        sz += __shfl_xor(sz, m, 32);
    }
    const int   nv    = num_voxels[p];
    const float invnv = 1.0f / (float)nv;
    const float mx = sx * invnv, my = sy * invnv, mz = sz * invnv;

    const float cx = (float)coors[p * 4 + 3] * 0.2f + 0.1f;     // VX, X_OFF
    const float cy = (float)coors[p * 4 + 2] * 0.2f + (-39.9f); // VY, Y_OFF

    const float msk = (lane < nv) ? 1.0f : 0.0f;
    frow[0]  = f.x * msk;
    frow[1]  = f.y * msk;
    frow[2]  = f.z * msk;
    frow[3]  = f.w * msk;
    frow[4]  = (f.x - mx) * msk;
    frow[5]  = (f.y - my) * msk;
    frow[6]  = (f.z - mz) * msk;
    frow[7]  = (f.x - cx) * msk;
    frow[8]  = (f.y - cy) * msk;
    frow[9]  = 0.0f;
    frow[10] = 0.0f;
    frow[11] = 0.0f;
}

// One 16-unit column tile of the pillar GEMM: two 16-row M-tiles x K=12 via
// three V_WMMA_F32_16X16X4_F32 k-steps each.
// A layout (f32 16x4): lanes 0-15 -> M=lane, K={k0,k0+1}; lanes 16-31 -> M=lane-16, K={k0+2,k0+3}.
// B layout (f32 4x16): mirrored with N=lane&15.
__device__ __forceinline__ void gemm_tile(
    const float* __restrict__ flw /* [32][KPAD] in LDS */,
    const float* __restrict__ Wl  /* [KPAD][UNITS] in LDS */,
    int nt, int lane, v8f& acc0, v8f& acc1)
{
    const int half = lane >> 4;
    const int lm   = lane & 15;
    const int u0   = nt * 16;
#pragma unroll
    for (int ks = 0; ks < 3; ++ks) {
        const int k0 = ks * 4 + 2 * half;
        v2f a0 = { flw[lm * KPAD + k0],        flw[lm * KPAD + k0 + 1] };
        v2f a1 = { flw[(16 + lm) * KPAD + k0], flw[(16 + lm) * KPAD + k0 + 1] };
        v2f b  = { Wl[k0 * UNITS + u0 + lm],   Wl[(k0 + 1) * UNITS + u0 + lm] };
        acc0 = __builtin_amdgcn_wmma_f32_16x16x4_f32(
            false, a0, false, b, (short)0, acc0, false, false);
        acc1 = __builtin_amdgcn_wmma_f32_16x16x4_f32(
            false, a1, false, b, (short)0, acc1, false, false);
    }
}

// ---------------- Pass A: partial BN statistics ----------------
__global__ __launch_bounds__(BLOCK) void pfn_stats_kernel(
    const float* __restrict__ features, const int* __restrict__ num_voxels,
    const int* __restrict__ coors, const float* __restrict__ W,
    float* __restrict__ partials /* [NBLK][128]: sum(64) | sumsq(64) */)
{
    __shared__ float Wl[KPAD * UNITS];
    __shared__ float fl[WAVES_PER_BLOCK][NPTS][KPAD];
    __shared__ float red[2][WAVES_PER_BLOCK][UNITS];

    const int tid  = threadIdx.x;
    const int wave = tid >> 5;
    const int lane = tid & 31;
    const int p    = blockIdx.x * WAVES_PER_BLOCK + wave;

    for (int i = tid; i < KPAD * UNITS; i += BLOCK)
        Wl[i] = (i < (CIN + 5) * UNITS) ? W[i] : 0.0f;

    const bool valid = (p < PILLARS);   // wave-uniform
    if (valid)
        pillar_feats_to_lds(features, num_voxels, coors, p, lane,
                            &fl[wave][lane][0]);
    __syncthreads();

    const int half = lane >> 4;
    const int lm   = lane & 15;

    if (valid) {
#pragma unroll
        for (int nt = 0; nt < 4; ++nt) {
            v8f a0 = {}, a1 = {};
            gemm_tile(&fl[wave][0][0], Wl, nt, lane, a0, a1);
            float ps = 0.0f, pq = 0.0f;
#pragma unroll
            for (int j = 0; j < 8; ++j) {
                ps += a0[j] + a1[j];
                pq += a0[j] * a0[j] + a1[j] * a1[j];
            }
            // lanes L and L+16 hold the same unit (nt*16 + lm): combine halves
            ps += __shfl_xor(ps, 16, 32);
            pq += __shfl_xor(pq, 16, 32);
            if (half == 0) {
                red[0][wave][nt * 16 + lm] = ps;
                red[1][wave][nt * 16 + lm] = pq;
            }
        }
    } else {
        if (half == 0) {
#pragma unroll
            for (int nt = 0; nt < 4; ++nt) {
                red[0][wave][nt * 16 + lm] = 0.0f;
                red[1][wave][nt * 16 + lm] = 0.0f;
            }
        }
    }
    __syncthreads();

    // Deterministic per-block reduction over the 8 waves.
    if (tid < 2 * UNITS) {
        const int which = tid >> 6;
        const int u     = tid & 63;
        float s = 0.0f;
#pragma unroll
        for (int w = 0; w < WAVES_PER_BLOCK; ++w) s += red[which][w][u];
        partials[(size_t)blockIdx.x * (2 * UNITS) + tid] = s;
    }
}

// ---------------- Pass B: finalize BN stats -> scale/bias ----------------
__global__ __launch_bounds__(128) void pfn_finalize_stats(
    const float* __restrict__ partials, const float* __restrict__ gamma,
    const float* __restrict__ beta, float* __restrict__ stats /* scale(64)|bias(64) */,
    int nblk)
{
    __shared__ float red[128];
    const int t = threadIdx.x;
    float s = 0.0f;
    for (int w = 0; w < nblk; ++w) s += partials[(size_t)w * 128 + t];
    red[t] = s;
    __syncthreads();
    if (t < 64) {
        const float cnt  = (float)PILLARS * (float)NPTS;
        const float mean = red[t] / cnt;
        float var = red[64 + t] / cnt - mean * mean;   // biased, as torch BN
        var = fmaxf(var, 0.0f);
        const float inv = rsqrtf(var + 1e-3f);
        const float sc  = inv * gamma[t];
        stats[t]      = sc;
        stats[64 + t] = beta[t] - mean * sc;
    }
}

// ---------------- Pass C: recompute GEMM, affine+relu+max, write out ----------------
__global__ __launch_bounds__(BLOCK) void pfn_output_kernel(
    const float* __restrict__ features, const int* __restrict__ num_voxels,
    const int* __restrict__ coors, const float* __restrict__ W,
    const float* __restrict__ stats, float* __restrict__ out /* [P][UNITS] */)
{
    __shared__ float Wl[KPAD * UNITS];
    __shared__ float fl[WAVES_PER_BLOCK][NPTS][KPAD];

    const int tid  = threadIdx.x;
    const int wave = tid >> 5;
    const int lane = tid & 31;
    const int p    = blockIdx.x * WAVES_PER_BLOCK + wave;

    for (int i = tid; i < KPAD * UNITS; i += BLOCK)
        Wl[i] = (i < (CIN + 5) * UNITS) ? W[i] : 0.0f;

    const bool valid = (p < PILLARS);   // wave-uniform
    if (valid)
        pillar_feats_to_lds(features, num_voxels, coors, p, lane,
                            &fl[wave][lane][0]);
    __syncthreads();

    if (!valid) return;

    const int half = lane >> 4;
    const int lm   = lane & 15;

#pragma unroll
    for (int nt = 0; nt < 4; ++nt) {
        v8f a0 = {}, a1 = {};
        gemm_tile(&fl[wave][0][0], Wl, nt, lane, a0, a1);

        const int   u  = nt * 16 + lm;
        const float sc = stats[u];
        const float bi = stats[64 + u];

        float m = -3.402823466e38f;
#pragma unroll
        for (int j = 0; j < 8; ++j) {
            m = fmaxf(m, fmaf(a0[j], sc, bi));
            m = fmaxf(m, fmaf(a1[j], sc, bi));
        }
        m = fmaxf(m, __shfl_xor(m, 16, 32));   // merge the two lane-halves (same unit)
        m = fmaxf(m, 0.0f);                    // relu(max) == max(relu)
        if (half == 0) out[(size_t)p * UNITS + u] = m;
    }
}

extern "C" void kernel_launch(void* const* d_in, const int* in_sizes, int n_in,
                              void* d_out, int out_size, void* d_ws, size_t ws_size,
                              hipStream_t stream)
{
    const float* features   = (const float*)d_in[0];
    const int*   num_voxels = (const int*)  d_in[1];
    const int*   coors      = (const int*)  d_in[2];
    const float* W          = (const float*)d_in[3];
    const float* gamma      = (const float*)d_in[4];
    const float* beta       = (const float*)d_in[5];
    float*       out        = (float*)d_out;

    float* partials = (float*)d_ws;                       // NBLK * 128 floats
    float* stats    = partials + (size_t)NBLK * 128;      // 128 floats (scale|bias)

    pfn_stats_kernel<<<NBLK, BLOCK, 0, stream>>>(features, num_voxels, coors, W,
                                                 partials);
    pfn_finalize_stats<<<1, 128, 0, stream>>>(partials, gamma, beta, stats, NBLK);
    pfn_output_kernel<<<NBLK, BLOCK, 0, stream>>>(features, num_voxels, coors, W,
                                                  stats, out);
}